// DeformableAttention_42640435315241
// MI455X (gfx1250) — compile-verified
//
#include <hip/hip_runtime.h>
#include <hip/hip_bf16.h>
#include <math.h>

#define B_   4
#define L1_  4096
#define L2_  21760
#define QD   256
#define VD   256
#define NH_  8
#define NL_  4
#define NP_  4
#define HD   32          // VD / NH

#define KDIM     256     // K of every GEMM here
#define LDS_PAD  260     // 256 + 4 dword row stride -> banks 4*ln+c, conflict-free

typedef __attribute__((ext_vector_type(2))) float v2f;
typedef __attribute__((ext_vector_type(8))) float v8f;
typedef __attribute__((ext_vector_type(4))) int   v4i;

#define WMMA_F32(A, Bf, C) \
    __builtin_amdgcn_wmma_f32_16x16x4_f32(false, (A), false, (Bf), (short)0, (C), false, false)

#if __has_builtin(__builtin_amdgcn_global_load_async_to_lds_b128) && \
    __has_builtin(__builtin_amdgcn_s_wait_asynccnt)
#define USE_ASYNC_LDS 1
typedef __attribute__((address_space(1))) v4i* gbl_v4i_ptr;   // global (AS1)
typedef __attribute__((address_space(3))) v4i* lds_v4i_ptr;   // LDS    (AS3)
#else
#define USE_ASYNC_LDS 0
#endif

// ---------------------------------------------------------------------------
// D = X @ W^T + bias  (X: MxK row-major, W: NxK row-major, D: MxN), K = 256.
// Block = 8 waves sharing one 64-wide N group: the 64x256 W tile (64 KB,
// contiguous rows) is staged into LDS once per block (global_load_async_to_
// lds_b128 + s_wait_asynccnt when available, else b128 copy). Each wave then
// computes a 16x64 tile: per k-step 1 global b64 (A, double-buffered one step
// ahead) + 4 ds_load_b64 (B, bank-conflict-free via row pad) + 4 WMMAs.
// Fragment layout (ISA 7.12.2, 32-bit):
//   A 16x4 : lane l -> A[l%16][2*(l/16)+j],  j=0,1  (2 VGPRs)
//   B 4x16 : lane l -> B[2*(l/16)+j][l%16]          (2 VGPRs)
//   C 16x16: lane l, vgpr i -> C[i+8*(l/16)][l%16]  (8 VGPRs)
// Requires: M % 128 == 0, N % 64 == 0. Optional per-row mask zeroing.
// ---------------------------------------------------------------------------
__global__ void gemm_xwt_bias_kernel(const float* __restrict__ X,
                                     const float* __restrict__ W,
                                     const float* __restrict__ bias,
                                     float* __restrict__ Y,
                                     int M, int N,
                                     const unsigned char* __restrict__ mask) {
    __shared__ float lw[64 * LDS_PAD];            // 65 KB: W tile, padded rows

    const int tid      = (int)threadIdx.x;        // 0..255
    const int blocksN  = N >> 6;
    const int tmGroup  = (int)blockIdx.x / blocksN;
    const int tn       = (int)blockIdx.x % blocksN;

    // ---- cooperative stage of W rows [tn*64, tn*64+64) x 256 into LDS ----
    {
        const float* wtile = W + (size_t)(tn * 64) * KDIM;   // contiguous 64 KB
#pragma unroll
        for (int j = 0; j < 16; ++j) {
            const int chunk = tid + j * 256;      // 4096 x 16B chunks
            const int row   = chunk >> 6;         // /64 chunks per row
            const int col4  = chunk & 63;         // float4 column
            const float* gsrc = wtile + (size_t)row * KDIM + col4 * 4;
            float*       ldst = &lw[row * LDS_PAD + col4 * 4];
#if USE_ASYNC_LDS
            __builtin_amdgcn_global_load_async_to_lds_b128(
                (gbl_v4i_ptr)gsrc, (lds_v4i_ptr)ldst, 0, 0);
#else
            *(float4*)ldst = *(const float4*)gsrc;
#endif
        }
#if USE_ASYNC_LDS
        __builtin_amdgcn_s_wait_asynccnt(0);
#endif
        __syncthreads();
    }

    const int wave = tid >> 5;                    // 0..7
    const int lane = tid & 31;
    const int lh   = lane >> 4;                   // 0 or 1
    const int ln   = lane & 15;
    const int tm   = tmGroup * 8 + wave;

    const float* xrow = X + (size_t)(tm * 16 + ln) * KDIM + 2 * lh;
    const float* lb0  = &lw[( 0 + ln) * LDS_PAD + 2 * lh];
    const float* lb1  = &lw[(16 + ln) * LDS_PAD + 2 * lh];
    const float* lb2  = &lw[(32 + ln) * LDS_PAD + 2 * lh];
    const float* lb3  = &lw[(48 + ln) * LDS_PAD + 2 * lh];

    v8f c0 = {}, c1 = {}, c2 = {}, c3 = {};

    v2f a = *(const v2f*)(xrow);                  // A fragment for k = 0
    for (int k = 0; k < KDIM - 4; k += 4) {
        v2f an = *(const v2f*)(xrow + k + 4);     // prefetch next A (global)
        v2f b0 = *(const v2f*)(lb0 + k);          // B fragments from LDS
        v2f b1 = *(const v2f*)(lb1 + k);
        v2f b2 = *(const v2f*)(lb2 + k);
        v2f b3 = *(const v2f*)(lb3 + k);
        c0 = WMMA_F32(a, b0, c0);
        c1 = WMMA_F32(a, b1, c1);
        c2 = WMMA_F32(a, b2, c2);
        c3 = WMMA_F32(a, b3, c3);
        a = an;
    }
    {   // last k-step
        const int k = KDIM - 4;
        v2f b0 = *(const v2f*)(lb0 + k);
        v2f b1 = *(const v2f*)(lb1 + k);
        v2f b2 = *(const v2f*)(lb2 + k);
        v2f b3 = *(const v2f*)(lb3 + k);
        c0 = WMMA_F32(a, b0, c0);
        c1 = WMMA_F32(a, b1, c1);
        c2 = WMMA_F32(a, b2, c2);
        c3 = WMMA_F32(a, b3, c3);
    }

#pragma unroll
    for (int j = 0; j < 4; ++j) {
        const v8f&  c    = (j == 0) ? c0 : (j == 1) ? c1 : (j == 2) ? c2 : c3;
        const int   col  = tn * 64 + j * 16 + ln;
        const float bcol = bias ? bias[col] : 0.0f;
#pragma unroll
        for (int i = 0; i < 8; ++i) {
            const int row = tm * 16 + i + 8 * lh;
            float v = c[i] + bcol;
            if (mask && mask[row]) v = 0.0f;
            Y[(size_t)row * N + col] = v;
        }
    }
}

// ---------------------------------------------------------------------------
// attn = softmax(query @ Wa^T + ba) over NL*NP = 16 logits per head.
// N = 128 fixed; each 16-wide tile column group == one head's softmax group,
// so the reduction is a cross-lane shuffle inside 16-lane halves.
// ---------------------------------------------------------------------------
__global__ void gemm_attn_softmax_kernel(const float* __restrict__ X,
                                         const float* __restrict__ W,
                                         const float* __restrict__ bias,
                                         float* __restrict__ Y,
                                         int M) {
    const int wave = (int)((blockIdx.x * blockDim.x + threadIdx.x) >> 5);
    const int lane = (int)(threadIdx.x & 31);
    const int tm = wave >> 3;          // tilesN = 128/16 = 8
    const int tn = wave & 7;           // == head index
    if (tm * 16 >= M) return;

    const int lh = lane >> 4;
    const int ln = lane & 15;

    const float* xrow = X + (size_t)(tm * 16 + ln) * KDIM + 2 * lh;
    const float* wrow = W + (size_t)(tn * 16 + ln) * KDIM + 2 * lh;

    v8f c = {};
    v2f a = *(const v2f*)(xrow);
    v2f b = *(const v2f*)(wrow);
    for (int k = 0; k < KDIM - 4; k += 4) {
        v2f an = *(const v2f*)(xrow + k + 4);
        v2f bn = *(const v2f*)(wrow + k + 4);
        c = WMMA_F32(a, b, c);
        a = an; b = bn;
    }
    c = WMMA_F32(a, b, c);

    const int   col  = tn * 16 + ln;
    const float bcol = bias[col];
#pragma unroll
    for (int i = 0; i < 8; ++i) {
        float v = c[i] + bcol;
        // max over the 16 lanes of this half (xor masks < 16 stay in-half)
        float mx = v;
        for (int m = 1; m < 16; m <<= 1) mx = fmaxf(mx, __shfl_xor(mx, m, 32));
        float e = __expf(v - mx);
        float s = e;
        for (int m = 1; m < 16; m <<= 1) s += __shfl_xor(s, m, 32);
        const int row = tm * 16 + i + 8 * lh;
        Y[(size_t)row * 128 + col] = e / s;
    }
}

// ---------------------------------------------------------------------------
// Deformable bilinear sampling + attention-weighted accumulation.
// One wave per (b, l1, h); lane = head channel (hd == 32 == wave size).
// All location/weight math is lane-uniform (SALU-friendly); every tap is one
// coalesced 128 B vector load of vproj[(b,flat), h*32 + lane], L2-resident.
// ---------------------------------------------------------------------------
__global__ void ms_deform_sample_kernel(const float* __restrict__ vproj,
                                        const float* __restrict__ off,
                                        const float* __restrict__ attn,
                                        const float* __restrict__ ref_w,
                                        const float* __restrict__ vvr,
                                        const int*   __restrict__ v_shape,
                                        const int*   __restrict__ v_start,
                                        float*       __restrict__ tmp) {
    const int wave = (int)((blockIdx.x * blockDim.x + threadIdx.x) >> 5);
    const int lane = (int)(threadIdx.x & 31);
    if (wave >= B_ * L1_ * NH_) return;

    const int h  = wave % NH_;
    const int bl = wave / NH_;          // b*L1 + l1
    const int b  = bl / L1_;

    const float4 rw       = *(const float4*)(ref_w + (size_t)bl * 4);
    const float* offrow   = off  + (size_t)bl * (NH_ * NL_ * NP_ * 2)
                                 + (size_t)h * (NL_ * NP_ * 2);
    const float* attnrow  = attn + (size_t)bl * (NH_ * NL_ * NP_)
                                 + (size_t)h * (NL_ * NP_);
    const float* vbase    = vproj + (size_t)b * L2_ * VD + h * HD + lane;

    float acc = 0.0f;
    for (int lvl = 0; lvl < NL_; ++lvl) {
        const int   Hi  = v_shape[lvl * 2 + 0];
        const int   Wi  = v_shape[lvl * 2 + 1];
        const int   st  = v_start[lvl];
        const float vrx = vvr[(b * NL_ + lvl) * 2 + 0];
        const float vry = vvr[(b * NL_ + lvl) * 2 + 1];
        for (int p = 0; p < NP_; ++p) {
            const int   idx = lvl * NP_ + p;
            const float ox  = offrow[idx * 2 + 0];
            const float oy  = offrow[idx * 2 + 1];
            const float aw  = attnrow[idx];

            const float locx = (rw.x + ox * 0.125f * rw.z) * vrx;
            const float locy = (rw.y + oy * 0.125f * rw.w) * vry;
            const float x = locx * (float)Wi - 0.5f;
            const float y = locy * (float)Hi - 0.5f;
            const float x0f = floorf(x), y0f = floorf(y);
            const float fx = x - x0f, fy = y - y0f;
            const int   x0 = (int)x0f, y0 = (int)y0f;
#pragma unroll
            for (int dy = 0; dy < 2; ++dy) {
#pragma unroll
                for (int dx = 0; dx < 2; ++dx) {
                    const int xi = x0 + dx;
                    const int yi = y0 + dy;
                    const float wx = dx ? fx : 1.0f - fx;
                    const float wy = dy ? fy : 1.0f - fy;
                    if (xi >= 0 && xi < Wi && yi >= 0 && yi < Hi) {
                        const int flat = st + yi * Wi + xi;
                        acc += aw * wx * wy * vbase[(size_t)flat * VD];
                    }
                }
            }
        }
    }
    tmp[(size_t)bl * VD + h * HD + lane] = acc;
}

// ---------------------------------------------------------------------------
extern "C" void kernel_launch(void* const* d_in, const int* in_sizes, int n_in,
                              void* d_out, int out_size, void* d_ws, size_t ws_size,
                              hipStream_t stream) {
    const float*         query   = (const float*)d_in[0];
    const float*         value   = (const float*)d_in[1];
    const int*           v_shape = (const int*)d_in[2];
    const unsigned char* v_mask  = (const unsigned char*)d_in[3];
    const int*           v_start = (const int*)d_in[4];
    const float*         v_valid = (const float*)d_in[5];
    const float*         ref_w   = (const float*)d_in[6];
    const float*         Wo      = (const float*)d_in[7];
    const float*         bo      = (const float*)d_in[8];
    const float*         Wa      = (const float*)d_in[9];
    const float*         ba      = (const float*)d_in[10];
    const float*         Wv      = (const float*)d_in[11];
    const float*         bv      = (const float*)d_in[12];
    const float*         Wout    = (const float*)d_in[13];
    const float*         bout    = (const float*)d_in[14];

    float* out  = (float*)d_out;                       // (B, L1, 256)
    float* attn = out + (size_t)B_ * L1_ * QD;         // (B, L1, 128)

    float* vproj = (float*)d_ws;                       // (B, L2, 256)  ~85 MB
    float* offs  = vproj + (size_t)B_ * L2_ * VD;      // (B, L1, 256)  ~16 MB
    float* tmp   = offs  + (size_t)B_ * L1_ * QD;      // (B, L1, 256)  ~16 MB

    const int BL2 = B_ * L2_;   // 87040  (divisible by 128)
    const int BL1 = B_ * L1_;   // 16384  (divisible by 128)

    // 1) value projection + mask:  vproj = mask0(value @ Wv^T + bv)
    {
        int blocks = (BL2 / 128) * (VD / 64);          // 680 * 4 = 2720
        gemm_xwt_bias_kernel<<<blocks, 256, 0, stream>>>(
            value, Wv, bv, vproj, BL2, VD, v_mask);
    }
    // 2a) sampling offsets: offs = query @ Wo^T + bo
    {
        int blocks = (BL1 / 128) * (QD / 64);          // 128 * 4 = 512
        gemm_xwt_bias_kernel<<<blocks, 256, 0, stream>>>(
            query, Wo, bo, offs, BL1, QD, nullptr);
    }
    // 2b) attention weights: attn = softmax(query @ Wa^T + ba)
    {
        int waves = (BL1 / 16) * (128 / 16);           // 8192
        gemm_attn_softmax_kernel<<<waves / 8, 256, 0, stream>>>(
            query, Wa, ba, attn, BL1);
    }
    // 3) deformable sampling + attention accumulation -> tmp
    {
        int waves = B_ * L1_ * NH_;                    // 131072
        ms_deform_sample_kernel<<<waves / 8, 256, 0, stream>>>(
            vproj, offs, attn, ref_w, v_valid, v_shape, v_start, tmp);
    }
    // 4) output projection: out = tmp @ Wout^T + bout
    {
        int blocks = (BL1 / 128) * (QD / 64);          // 512
        gemm_xwt_bias_kernel<<<blocks, 256, 0, stream>>>(
            tmp, Wout, bout, out, BL1, QD, nullptr);
    }
}